// FCPairedLayer_76836964926031
// MI455X (gfx1250) — compile-verified
//
#include <hip/hip_runtime.h>

// ---------------------------------------------------------------------------
// FCPairedLayer on MI455X (gfx1250)
//
// Decomposition:
//   G0[r] = X[r]   @ W1[:,   0: 64].T + b1    (term at i,   always)
//   G1[r] = X[r]   @ W1[:,  64:128].T         (term at j,   always)
//   G2[r] = X[r]   @ W1[:, 128:192].T         (used at i-1, mask mu)
//   G3[r] = X[r]   @ W1[:, 192:256].T         (used at j+1, mask mu)
//   G4[r] = X[r]   @ W1[:, 256:320].T         (used at i+1, mask md)
//   G5[r] = X[r]   @ W1[:, 320:384].T         (used at j-1, mask md)
// with mu = (i>=1 && j<=N-2), md = (i<=N-2 && j>=1).
// y[b,i,j] = sum_h W2[h]*relu(G0[i]+G1[j]+mu*(G2[i-1]+G3[j+1])+md*(G4[i+1]+G5[j-1])) + b2
// ---------------------------------------------------------------------------

typedef __attribute__((ext_vector_type(2))) float v2f;
typedef __attribute__((ext_vector_type(4))) float v4f;
typedef __attribute__((ext_vector_type(8))) float v8f;

#define NSEQ 384
#define BATCH 2
#define ROWS (BATCH * NSEQ)   // 768
#define CIN 64
#define HID 128
#define LPITCH 132            // LDS pitch in floats: 132 % 64 == 4 -> conflict-free

// ---------------------------------------------------------------------------
// Stage 1: six [768x64] x [64x128] GEMMs via V_WMMA_F32_16X16X4_F32.
// One wave32 per 16x16 output tile; 16 chained WMMAs across K=64.
// A (16x4 f32):  lane l<16 holds A[m=l][k0+0], A[m=l][k0+1] in (v0,v1);
//                lane l>=16 holds A[m=l-16][k0+2], A[m=l-16][k0+3].
// B (4x16 f32):  mirrored: lane group selects K pair, lane&15 selects N.
// C/D (16x16):   vgpr r, lanes 0-15 -> M=r; lanes 16-31 -> M=r+8; N=lane&15.
// ---------------------------------------------------------------------------
__global__ __launch_bounds__(32) void fcpair_stage1_wmma(
    const float* __restrict__ X,   // [ROWS, 64]
    const float* __restrict__ W1,  // [128, 384] row-major
    const float* __restrict__ b1,  // [128]
    float* __restrict__ G)         // [6, ROWS, 128]
{
    const int s    = blockIdx.z;        // slice 0..5
    const int rt   = blockIdx.x;        // row tile 0..47
    const int ct   = blockIdx.y;        // col tile 0..7
    const int lane = threadIdx.x;
    const int half = lane >> 4;
    const int l15  = lane & 15;

    const int arow = rt * 16 + l15;     // A-matrix M index for this lane
    const int hcol = ct * 16 + l15;     // B-matrix N index == hidden channel

    const float* aptr = X  + arow * CIN + 2 * half;
    const float* bptr = W1 + hcol * (6 * CIN) + s * CIN + 2 * half;

    v8f acc = {};
#pragma unroll
    for (int k0 = 0; k0 < CIN; k0 += 4) {
        v2f a  = *(const v2f*)(aptr + k0);
        v2f bb = *(const v2f*)(bptr + k0);
        acc = __builtin_amdgcn_wmma_f32_16x16x4_f32(
            /*neg_a=*/false, a, /*neg_b=*/false, bb,
            /*c_mod=*/(short)0, acc, /*reuse_a=*/false, /*reuse_b=*/false);
    }

    const float badd = (s == 0) ? b1[hcol] : 0.0f;
    float* gout = G + (size_t)s * ROWS * HID;
#pragma unroll
    for (int r = 0; r < 8; ++r) {
        const int grow = rt * 16 + half * 8 + r;
        gout[grow * HID + hcol] = acc[r] + badd;
    }
}

// ---------------------------------------------------------------------------
// Stage 2: pairwise combine + relu + dot(W2) + b2.
// Block = 256 threads, tile = 16 i x 32 j (2 pairs/thread).
// LDS pitch 132 floats -> 16 lanes hit 16 distinct 4-bank groups (b128 reads).
// ---------------------------------------------------------------------------
__global__ __launch_bounds__(256) void fcpair_stage2_pair(
    const float* __restrict__ G,   // [6, ROWS, 128]
    const float* __restrict__ W2,  // [128]
    const float* __restrict__ b2,  // [1]
    float* __restrict__ out)       // [BATCH, N, N]
{
    __shared__ alignas(16) float rowS[3 * 16 * LPITCH]; // G0/G2/G4 at i / i-1 / i+1
    __shared__ alignas(16) float colS[3 * 32 * LPITCH]; // G1/G3/G5 at j / j+1 / j-1
    __shared__ alignas(16) float w2s[HID];

    const int b   = blockIdx.z;
    const int i0  = blockIdx.y * 16;
    const int j0  = blockIdx.x * 32;
    const int tid = threadIdx.x;

    const float* G0 = G + (size_t)0 * ROWS * HID;
    const float* G1 = G + (size_t)1 * ROWS * HID;
    const float* G2 = G + (size_t)2 * ROWS * HID;
    const float* G3 = G + (size_t)3 * ROWS * HID;
    const float* G4 = G + (size_t)4 * ROWS * HID;
    const float* G5 = G + (size_t)5 * ROWS * HID;

    // ---- load row side: 3 arrays x 16 rows x 32 quads --------------------
    for (int idx = tid; idx < 3 * 16 * 32; idx += 256) {
        const int arr = idx / (16 * 32);
        const int rem = idx - arr * (16 * 32);
        const int ii  = rem >> 5;
        const int q   = rem & 31;
        const int i   = i0 + ii;
        const int ri  = b * NSEQ + i;
        const float* src;
        if (arr == 0)      src = G0 + (size_t)ri * HID;
        else if (arr == 1) src = G2 + (size_t)((i >= 1)        ? ri - 1 : ri) * HID;
        else               src = G4 + (size_t)((i <= NSEQ - 2) ? ri + 1 : ri) * HID;
        *(v4f*)&rowS[(arr * 16 + ii) * LPITCH + q * 4] = *(const v4f*)(src + q * 4);
    }
    // ---- load col side: 3 arrays x 32 rows x 32 quads --------------------
    for (int idx = tid; idx < 3 * 32 * 32; idx += 256) {
        const int arr = idx / (32 * 32);
        const int rem = idx - arr * (32 * 32);
        const int jj  = rem >> 5;
        const int q   = rem & 31;
        const int j   = j0 + jj;
        const int rj  = b * NSEQ + j;
        const float* src;
        if (arr == 0)      src = G1 + (size_t)rj * HID;
        else if (arr == 1) src = G3 + (size_t)((j <= NSEQ - 2) ? rj + 1 : rj) * HID;
        else               src = G5 + (size_t)((j >= 1)        ? rj - 1 : rj) * HID;
        *(v4f*)&colS[(arr * 32 + jj) * LPITCH + q * 4] = *(const v4f*)(src + q * 4);
    }
    if (tid < HID) w2s[tid] = W2[tid];
    __syncthreads();

    // ---- compute: thread owns (ii, jc) and (ii, jc+16) -------------------
    const int ii = tid >> 4;
    const int jc = tid & 15;
    const int i  = i0 + ii;
    const int jA = j0 + jc;
    const int jB = jA + 16;

    const float mi  = (i >= 1)        ? 1.0f : 0.0f;
    const float mip = (i <= NSEQ - 2) ? 1.0f : 0.0f;
    const float muA = mi  * ((jA <= NSEQ - 2) ? 1.0f : 0.0f);
    const float mdA = mip * ((jA >= 1)        ? 1.0f : 0.0f);
    const float muB = mi  * ((jB <= NSEQ - 2) ? 1.0f : 0.0f);
    const float mdB = mip * ((jB >= 1)        ? 1.0f : 0.0f);

    const float* r0p = &rowS[(0 * 16 + ii) * LPITCH];
    const float* r1p = &rowS[(1 * 16 + ii) * LPITCH];
    const float* r2p = &rowS[(2 * 16 + ii) * LPITCH];
    const float* a0p = &colS[(0 * 32 + jc) * LPITCH];
    const float* a1p = &colS[(1 * 32 + jc) * LPITCH];
    const float* a2p = &colS[(2 * 32 + jc) * LPITCH];
    const float* b0p = &colS[(0 * 32 + jc + 16) * LPITCH];
    const float* b1p = &colS[(1 * 32 + jc + 16) * LPITCH];
    const float* b2p = &colS[(2 * 32 + jc + 16) * LPITCH];

    float accA = 0.0f, accB = 0.0f;
#pragma unroll 4
    for (int h = 0; h < HID; h += 4) {
        const v4f r0 = *(const v4f*)(r0p + h);
        const v4f r1 = *(const v4f*)(r1p + h);
        const v4f r2 = *(const v4f*)(r2p + h);
        const v4f a0 = *(const v4f*)(a0p + h);
        const v4f a1 = *(const v4f*)(a1p + h);
        const v4f a2 = *(const v4f*)(a2p + h);
        const v4f c0 = *(const v4f*)(b0p + h);
        const v4f c1 = *(const v4f*)(b1p + h);
        const v4f c2 = *(const v4f*)(b2p + h);
        const v4f w  = *(const v4f*)&w2s[h];
#pragma unroll
        for (int e = 0; e < 4; ++e) {
            float sA = r0[e] + a0[e] + muA * (r1[e] + a1[e]) + mdA * (r2[e] + a2[e]);
            accA = fmaf(w[e], fmaxf(sA, 0.0f), accA);
            float sB = r0[e] + c0[e] + muB * (r1[e] + c1[e]) + mdB * (r2[e] + c2[e]);
            accB = fmaf(w[e], fmaxf(sB, 0.0f), accB);
        }
    }
    const float bias2 = b2[0];
    const size_t base = ((size_t)(b * NSEQ + i)) * NSEQ;
    out[base + jA] = accA + bias2;
    out[base + jB] = accB + bias2;
}

// ---------------------------------------------------------------------------
extern "C" void kernel_launch(void* const* d_in, const int* in_sizes, int n_in,
                              void* d_out, int out_size, void* d_ws, size_t ws_size,
                              hipStream_t stream) {
    (void)in_sizes; (void)n_in; (void)out_size; (void)ws_size;
    const float* x  = (const float*)d_in[0];  // [2,384,64]
    const float* W1 = (const float*)d_in[1];  // [128,384]
    const float* b1 = (const float*)d_in[2];  // [128]
    const float* W2 = (const float*)d_in[3];  // [128]
    const float* b2 = (const float*)d_in[4];  // [1]
    float* out = (float*)d_out;               // [2,384,384]
    float* G   = (float*)d_ws;                // 6*768*128 floats = 2.25 MB

    // Stage 1: 48 row-tiles x 8 col-tiles x 6 slices, one wave each.
    fcpair_stage1_wmma<<<dim3(48, 8, 6), 32, 0, stream>>>(x, W1, b1, G);
    // Stage 2: 12 j-tiles x 24 i-tiles x 2 batches, 256 threads each.
    fcpair_stage2_pair<<<dim3(12, 24, 2), 256, 0, stream>>>(G, W2, b2, out);
}